// HCHA_74509092651627
// MI455X (gfx1250) — compile-verified
//
#include <hip/hip_runtime.h>
#include <hip/hip_bf16.h>

// ---------------------------------------------------------------------------
// HCHA hypergraph conv forward, MI455X (gfx1250, wave32).
//   Per launch: build CSR adjacency (v->e and e->v) once, then 3 layers of:
//     msg = h @ W                          (fp32 WMMA 16x16x4)
//     xe[h] = (sum_{v in adj_e[h]} msg[v]) / deg_e[h]          (gather, no atomics)
//     hv[v] = ELU((sum_{e in adj_v[v]} xe[e]) / deg_v[v] + b)  (gather, fused epilogue)
// Rationale: gather form removes ~1B f32 atomics/launch; msg+xe (102MB) stay
// resident in the 192MB L2, so the pass runs at L2 bandwidth.
// ---------------------------------------------------------------------------

#define N_NODES  50000
#define N_HEDGES 50000
#define NNZ      800000
#define F_IN     256
#define HID      256
#define OUT_C    128

typedef float v2f __attribute__((ext_vector_type(2)));
typedef float v8f __attribute__((ext_vector_type(8)));

// ---------------------------------------------------------------------------
// Zero-fill (graph-capture safe; used for int counters/cursors only)
// ---------------------------------------------------------------------------
__global__ void hcha_zero_kernel(float4* __restrict__ p, long n4) {
    long i      = (long)blockIdx.x * blockDim.x + threadIdx.x;
    long stride = (long)gridDim.x * blockDim.x;
    for (; i < n4; i += stride) p[i] = make_float4(0.f, 0.f, 0.f, 0.f);
}

// ---------------------------------------------------------------------------
// Degree counting (int atomics, 1.6M total — negligible)
// ---------------------------------------------------------------------------
__global__ void hcha_deg_kernel(const int* __restrict__ vi, const int* __restrict__ hi,
                                int* __restrict__ cnt_v, int* __restrict__ cnt_e) {
    int e = blockIdx.x * blockDim.x + threadIdx.x;
    if (e >= NNZ) return;
    __hip_atomic_fetch_add(cnt_v + vi[e], 1, __ATOMIC_RELAXED, __HIP_MEMORY_SCOPE_AGENT);
    __hip_atomic_fetch_add(cnt_e + hi[e], 1, __ATOMIC_RELAXED, __HIP_MEMORY_SCOPE_AGENT);
}

__global__ void hcha_invdeg_kernel(const int* __restrict__ cnt, float* __restrict__ inv, int n) {
    int i = blockIdx.x * blockDim.x + threadIdx.x;
    if (i < n) inv[i] = 1.0f / fmaxf((float)cnt[i], 1.0f);
}

// ---------------------------------------------------------------------------
// Single-workgroup exclusive scan (n = 50000: 49 chunks of 1024, trivial cost)
// offsets[0..n], offsets[n] = total
// ---------------------------------------------------------------------------
__global__ __launch_bounds__(1024) void hcha_scan_kernel(const int* __restrict__ counts,
                                                         int* __restrict__ offsets, int n) {
    __shared__ int lds[1024];
    __shared__ int carry_s;
    const int tid = threadIdx.x;
    if (tid == 0) carry_s = 0;
    __syncthreads();
    for (int base = 0; base < n; base += 1024) {
        const int i = base + tid;
        const int v = (i < n) ? counts[i] : 0;
        lds[tid] = v;
        __syncthreads();
        #pragma unroll
        for (int off = 1; off < 1024; off <<= 1) {
            int t = (tid >= off) ? lds[tid - off] : 0;
            __syncthreads();
            lds[tid] += t;
            __syncthreads();
        }
        const int incl = lds[tid];
        const int c    = carry_s;
        if (i < n) offsets[i] = c + incl - v;   // exclusive
        __syncthreads();
        if (tid == 1023) carry_s = c + incl;    // chunk total
        __syncthreads();
    }
    if (tid == 0) offsets[n] = carry_s;
}

// ---------------------------------------------------------------------------
// CSR fill: adj_e (edges grouped by hedge, storing v), adj_v (grouped by node,
// storing h). One returning int atomic per edge per direction.
// ---------------------------------------------------------------------------
__global__ void hcha_fill_adj(const int* __restrict__ vi, const int* __restrict__ hi,
                              const int* __restrict__ off_v, const int* __restrict__ off_e,
                              int* __restrict__ cur_v, int* __restrict__ cur_e,
                              int* __restrict__ adj_v, int* __restrict__ adj_e) {
    int e = blockIdx.x * blockDim.x + threadIdx.x;
    if (e >= NNZ) return;
    const int v = vi[e], h = hi[e];
    const int pe = __hip_atomic_fetch_add(cur_e + h, 1, __ATOMIC_RELAXED, __HIP_MEMORY_SCOPE_AGENT);
    adj_e[off_e[h] + pe] = v;
    const int pv = __hip_atomic_fetch_add(cur_v + v, 1, __ATOMIC_RELAXED, __HIP_MEMORY_SCOPE_AGENT);
    adj_v[off_v[v] + pv] = h;
}

// ---------------------------------------------------------------------------
// GEMM: D[M,N] = A[M,K] @ B[K,N], fp32, via V_WMMA_F32_16X16X4_F32.
// One wave per 16x64 output strip; A fragment reused across 4 WMMAs.
// M%16==0, N%64==0 -> no divergence, EXEC all-ones as WMMA requires.
// ---------------------------------------------------------------------------
__global__ __launch_bounds__(32) void hcha_wmma_gemm_f32(
    const float* __restrict__ A, const float* __restrict__ B,
    float* __restrict__ D, int K, int N)
{
    const int lane = threadIdx.x & 31;
    const int half = lane >> 4;
    const int l15  = lane & 15;
    const int m0   = blockIdx.x << 4;
    const int n0   = blockIdx.y << 6;

    v8f acc0 = {}; v8f acc1 = {}; v8f acc2 = {}; v8f acc3 = {};

    const float* aptr  = A + (size_t)(m0 + l15) * K + (half ? 2 : 0);
    const float* bbase = B + (size_t)(half ? 2 : 0) * N + n0 + l15;

    #pragma unroll 4
    for (int k = 0; k < K; k += 4) {
        const float2 av = *(const float2*)(aptr + k);
        v2f a; a.x = av.x; a.y = av.y;

        const float* bp = bbase + (size_t)k * N;
        v2f b0, b1, b2, b3;
        b0.x = bp[0];  b0.y = bp[N];
        b1.x = bp[16]; b1.y = bp[N + 16];
        b2.x = bp[32]; b2.y = bp[N + 32];
        b3.x = bp[48]; b3.y = bp[N + 48];

        acc0 = __builtin_amdgcn_wmma_f32_16x16x4_f32(false, a, false, b0, (short)0, acc0, false, false);
        acc1 = __builtin_amdgcn_wmma_f32_16x16x4_f32(false, a, false, b1, (short)0, acc1, false, false);
        acc2 = __builtin_amdgcn_wmma_f32_16x16x4_f32(false, a, false, b2, (short)0, acc2, false, false);
        acc3 = __builtin_amdgcn_wmma_f32_16x16x4_f32(false, a, false, b3, (short)0, acc3, false, false);
    }

    const int srow = m0 + (half << 3);
    const int scol = n0 + l15;
    float* drow = D + (size_t)srow * N + scol;
    #pragma unroll
    for (int r = 0; r < 8; ++r) {
        drow[(size_t)r * N +  0] = acc0[r];
        drow[(size_t)r * N + 16] = acc1[r];
        drow[(size_t)r * N + 32] = acc2[r];
        drow[(size_t)r * N + 48] = acc3[r];
    }
}

// ---------------------------------------------------------------------------
// CSR gather:  dst[r] = reduce_{j in adj[off[r]..off[r+1])} src[adj[j]]
//              * inv_deg[r]  (+ bias, + ELU)      -- no atomics, plain stores.
// Block = 256 threads, each thread owns 4 channels (float4); several rows per
// block. Next-neighbor row prefetched (global_prefetch_b8) since addresses are
// data-dependent and defeat HW prefetch.
// ---------------------------------------------------------------------------
__global__ __launch_bounds__(256) void hcha_gather(
    const float* __restrict__ src, const int* __restrict__ adj,
    const int* __restrict__ off, const float* __restrict__ inv_deg,
    const float* __restrict__ bias, float* __restrict__ dst,
    int nRows, int C, int do_elu)
{
    const int tpr = C >> 2;                       // threads per row (64 or 32)
    const int row = blockIdx.x * (256 / tpr) + threadIdx.x / tpr;
    if (row >= nRows) return;
    const int c4  = (threadIdx.x % tpr) << 2;

    const int beg = off[row];
    const int end = off[row + 1];

    float ax = 0.f, ay = 0.f, az = 0.f, aw = 0.f;
    for (int j = beg; j < end; ++j) {
        const int r = adj[j];
        if (j + 1 < end) {
            const int rn = adj[j + 1];
            __builtin_prefetch(src + (size_t)rn * C + c4, 0, 0);
        }
        const float4 m = *(const float4*)(src + (size_t)r * C + c4);
        ax += m.x; ay += m.y; az += m.z; aw += m.w;
    }

    const float s = inv_deg[row];
    float ox = ax * s, oy = ay * s, oz = az * s, ow = aw * s;
    if (bias) {
        const float4 b = *(const float4*)(bias + c4);
        ox += b.x; oy += b.y; oz += b.z; ow += b.w;
    }
    if (do_elu) {
        ox = (ox > 0.f) ? ox : (expf(ox) - 1.f);
        oy = (oy > 0.f) ? oy : (expf(oy) - 1.f);
        oz = (oz > 0.f) ? oz : (expf(oz) - 1.f);
        ow = (ow > 0.f) ? ow : (expf(ow) - 1.f);
    }
    float4 o; o.x = ox; o.y = oy; o.z = oz; o.w = ow;
    *(float4*)(dst + (size_t)row * C + c4) = o;
}

// ---------------------------------------------------------------------------
// Host-side orchestration
// ---------------------------------------------------------------------------
static void hcha_run_layer(const float* hin, const float* W, const float* bias,
                           float* msg, float* xe,
                           const int* adj_e, const int* off_e, const float* inv_de,
                           const int* adj_v, const int* off_v, const float* inv_dv,
                           float* hout, int N, int do_elu, hipStream_t stream)
{
    // GEMM: msg = hin @ W   (K = 256 for all layers)
    dim3 ggrid(N_NODES / 16, N / 64);
    hcha_wmma_gemm_f32<<<ggrid, 32, 0, stream>>>(hin, W, msg, HID, N);

    const int tpr = N >> 2;
    const int rpb = 256 / tpr;

    // v -> e gather: xe = mean of incident node messages
    hcha_gather<<<(N_HEDGES + rpb - 1) / rpb, 256, 0, stream>>>(
        msg, adj_e, off_e, inv_de, nullptr, xe, N_HEDGES, N, 0);

    // e -> v gather: hout = ELU(mean of incident hedge feats + bias)
    hcha_gather<<<(N_NODES + rpb - 1) / rpb, 256, 0, stream>>>(
        xe, adj_v, off_v, inv_dv, bias, hout, N_NODES, N, do_elu);
}

extern "C" void kernel_launch(void* const* d_in, const int* in_sizes, int n_in,
                              void* d_out, int out_size, void* d_ws, size_t ws_size,
                              hipStream_t stream) {
    const float* x   = (const float*)d_in[0];
    const int*   ei  = (const int*)d_in[1];   // [2, NNZ] int32 (harness contract)
    const float* W1  = (const float*)d_in[3];
    const float* b1  = (const float*)d_in[4];
    const float* W2  = (const float*)d_in[5];
    const float* b2  = (const float*)d_in[6];
    const float* W3  = (const float*)d_in[7];
    const float* b3  = (const float*)d_in[8];
    const int* vi = ei;        // row 0
    const int* hi = ei + NNZ;  // row 1

    // ---------------- workspace layout (~162 MB) ----------------
    char* ws = (char*)d_ws;
    const size_t big = (size_t)N_NODES * HID * sizeof(float);   // 51.2 MB
    float* msg = (float*)(ws + 0 * big);
    float* xe  = (float*)(ws + 1 * big);
    float* h2  = (float*)(ws + 2 * big);
    char*  p   = ws + 3 * big;
    int* cnt_v = (int*)p;              p += (size_t)N_NODES  * 4;
    int* cnt_e = (int*)p;              p += (size_t)N_HEDGES * 4;
    int* cur_v = (int*)p;              p += (size_t)N_NODES  * 4;
    int* cur_e = (int*)p;              p += (size_t)N_HEDGES * 4;
    int* off_v = (int*)p;              p += (size_t)(N_NODES  + 4) * 4;
    int* off_e = (int*)p;              p += (size_t)(N_HEDGES + 4) * 4;
    int* adj_v = (int*)p;              p += (size_t)NNZ * 4;
    int* adj_e = (int*)p;              p += (size_t)NNZ * 4;
    float* inv_dv = (float*)p;         p += (size_t)N_NODES  * 4;
    float* inv_de = (float*)p;         p += (size_t)N_HEDGES * 4;

    // ---------------- build CSR adjacency (once per launch) ----------------
    // zero counters + cursors (contiguous: cnt_v|cnt_e|cur_v|cur_e)
    hcha_zero_kernel<<<256, 256, 0, stream>>>((float4*)cnt_v,
        (long)(2 * (N_NODES + N_HEDGES)) / 4);
    hcha_deg_kernel<<<(NNZ + 255) / 256, 256, 0, stream>>>(vi, hi, cnt_v, cnt_e);
    hcha_invdeg_kernel<<<(N_NODES  + 255) / 256, 256, 0, stream>>>(cnt_v, inv_dv, N_NODES);
    hcha_invdeg_kernel<<<(N_HEDGES + 255) / 256, 256, 0, stream>>>(cnt_e, inv_de, N_HEDGES);
    hcha_scan_kernel<<<1, 1024, 0, stream>>>(cnt_v, off_v, N_NODES);
    hcha_scan_kernel<<<1, 1024, 0, stream>>>(cnt_e, off_e, N_HEDGES);
    hcha_fill_adj<<<(NNZ + 255) / 256, 256, 0, stream>>>(
        vi, hi, off_v, off_e, cur_v, cur_e, adj_v, adj_e);

    // ---------------- layers ----------------
    hcha_run_layer(x,  W1, b1, msg, xe, adj_e, off_e, inv_de, adj_v, off_v, inv_dv,
                   h2, HID, 1, stream);
    hcha_run_layer(h2, W2, b2, msg, xe, adj_e, off_e, inv_de, adj_v, off_v, inv_dv,
                   h2, HID, 1, stream);
    hcha_run_layer(h2, W3, b3, msg, xe, adj_e, off_e, inv_de, adj_v, off_v, inv_dv,
                   (float*)d_out, OUT_C, 0, stream);
}